// CoarseNetIter_35210141893224
// MI455X (gfx1250) — compile-verified
//
#include <hip/hip_runtime.h>

// ---------------- problem constants (match reference) ----------------
#define Bz   8
#define Hh   512
#define Ww   512
#define S1c  2048
#define S2c  8192
#define DEGc 8
#define N1c  (Bz * S1c)   // 16384
#define N2c  (Bz * S2c)   // 65536
#define E1c  (N1c * DEGc) // 131072
#define E2c  (N2c * DEGc) // 524288
static const size_t HWc = (size_t)Hh * Ww;

typedef float v2f __attribute__((ext_vector_type(2)));
typedef float v8f __attribute__((ext_vector_type(8)));

// ---------------- tiny utility kernels ----------------
__global__ void zero_i32_k(int* __restrict__ p, int n) {
  int i = blockIdx.x * blockDim.x + threadIdx.x;
  if (i < n) p[i] = 0;
}
__global__ void zero_f32_k(float* __restrict__ p, int n) {
  int i = blockIdx.x * blockDim.x + threadIdx.x;
  if (i < n) p[i] = 0.f;
}
__global__ void copy_i32_k(const int* __restrict__ a, int* __restrict__ b, int n) {
  int i = blockIdx.x * blockDim.x + threadIdx.x;
  if (i < n) b[i] = a[i];
}

// xin[i] = {x,y,z,mask}
__global__ void build_xin_k(const float* __restrict__ x, const float* __restrict__ m,
                            float* __restrict__ xin, int n) {
  int i = blockIdx.x * blockDim.x + threadIdx.x;
  if (i >= n) return;
  xin[i * 4 + 0] = x[i * 3 + 0];
  xin[i * 4 + 1] = x[i * 3 + 1];
  xin[i * 4 + 2] = x[i * 3 + 2];
  xin[i * 4 + 3] = m[i];
}

// ---------------- degree + CSR construction ----------------
__global__ void count_deg_k(const int* __restrict__ dst, int E, int* __restrict__ cnt) {
  int e = blockIdx.x * blockDim.x + threadIdx.x;
  if (e < E) atomicAdd(&cnt[dst[e]], 1);
}

// dis = rsqrt(deg + 2)  (improved self-loop weight 2);  invc = 1/(deg + 1)
__global__ void deg_fin_k(const int* __restrict__ cnt, float* __restrict__ dis,
                          float* __restrict__ invc, int n) {
  int i = blockIdx.x * blockDim.x + threadIdx.x;
  if (i >= n) return;
  float c = (float)cnt[i];
  dis[i]  = rsqrtf(c + 2.f);
  invc[i] = 1.f / (c + 1.f);
}

// single-block exclusive scan (n up to 64K; 1024-wide Hillis-Steele chunks)
__global__ void scan_excl_k(const int* __restrict__ cnt, int* __restrict__ rowp, int n) {
  __shared__ int sh[1024];
  __shared__ int carry;
  int tid = threadIdx.x;
  if (tid == 0) carry = 0;
  __syncthreads();
  for (int base = 0; base < n; base += 1024) {
    int v = (base + tid < n) ? cnt[base + tid] : 0;
    sh[tid] = v;
    __syncthreads();
    for (int off = 1; off < 1024; off <<= 1) {
      int t = (tid >= off) ? sh[tid - off] : 0;
      __syncthreads();
      sh[tid] += t;
      __syncthreads();
    }
    if (base + tid < n) rowp[base + tid] = carry + sh[tid] - v;
    __syncthreads();
    if (tid == 0) carry += sh[1023];
    __syncthreads();
  }
  if (tid == 0) rowp[n] = carry;
}

__global__ void csr_place_k(const int* __restrict__ dst, int E, int* __restrict__ cursor,
                            int* __restrict__ csre) {
  int e = blockIdx.x * blockDim.x + threadIdx.x;
  if (e >= E) return;
  int p = atomicAdd(&cursor[dst[e]], 1);
  csre[p] = e;
}

// canonicalize each node's edge list (sort by edge id) -> deterministic float sums
__global__ void csr_sort_k(const int* __restrict__ rowp, int* __restrict__ csre, int n) {
  int i = blockIdx.x * blockDim.x + threadIdx.x;
  if (i >= n) return;
  int a = rowp[i], b = rowp[i + 1];
  for (int x = a + 1; x < b; ++x) {
    int v = csre[x];
    int y = x - 1;
    while (y >= a && csre[y] > v) { csre[y + 1] = csre[y]; --y; }
    csre[y + 1] = v;
  }
}

// ---------------- dense matmul via FP32 WMMA (16x16x4) ----------------
// One wave computes one 16x16 tile of Out = A[n,ci] * W[ci,co].
// Fragment layouts per CDNA5 ISA:
//   A 16x4: lanes 0-15 -> {K=k,k+1} of row (lane); lanes 16-31 -> {K=k+2,k+3}
//   B 4x16: lane&15 = column; half selects K pair (symmetric to A)
//   C/D 16x16: VGPR r -> row r + 8*half, col = lane&15
__global__ __launch_bounds__(32) void matmul_wmma_k(const float* __restrict__ A,
                                                    const float* __restrict__ W,
                                                    float* __restrict__ Out,
                                                    int n, int ci, int co) {
  int tilesN = co >> 4;
  int tile   = blockIdx.x;
  int tm     = tile / tilesN;
  int tn     = tile - tm * tilesN;
  int lane   = threadIdx.x;
  int half   = lane >> 4;
  int l      = lane & 15;
  int row    = tm * 16 + l;   // A row for this lane
  int col    = tn * 16 + l;   // B/D column for this lane
  v8f acc = {0.f, 0.f, 0.f, 0.f, 0.f, 0.f, 0.f, 0.f};
  const size_t arow = (size_t)row * ci;
  for (int k = 0; k < ci; k += 4) {
    int ka = k + half * 2;
    v2f a, b;
    a.x = A[arow + ka];
    a.y = A[arow + ka + 1];
    b.x = W[(size_t)ka * co + col];
    b.y = W[(size_t)(ka + 1) * co + col];
    acc = __builtin_amdgcn_wmma_f32_16x16x4_f32(false, a, false, b, (short)0, acc,
                                                false, false);
  }
#pragma unroll
  for (int r = 0; r < 8; ++r) {
    int m = tm * 16 + r + half * 8;
    Out[(size_t)m * co + col] = acc[r];
  }
}

// scalar matmul for the final 32->3 projection (co not a multiple of 16)
__global__ void matmul_small_k(const float* __restrict__ A, const float* __restrict__ W,
                               float* __restrict__ Out, int n, int ci, int co) {
  int idx = blockIdx.x * blockDim.x + threadIdx.x;
  if (idx >= n * co) return;
  int node = idx / co, c = idx - node * co;
  float s = 0.f;
  for (int k = 0; k < ci; ++k) s += A[(size_t)node * ci + k] * W[k * co + c];
  Out[idx] = s;
}

// ---------------- GCN aggregation (CSR gather, fused gate + residual) ----------------
// agg_k[i] = xw_k[i]*2*dis[i]^2 + sum_e dis[src]*dis[i]*xw_k[src];  h = gate(agg/(cnt+1)) (+resid)
__global__ void aggregate_k(const float* __restrict__ xw1, const float* __restrict__ xw2,
                            const int* __restrict__ rowp, const int* __restrict__ csre,
                            const int* __restrict__ src, const float* __restrict__ dis,
                            const float* __restrict__ invc, const float* __restrict__ resid,
                            float* __restrict__ out, int n, int co, int gated) {
  int idx = blockIdx.x * blockDim.x + threadIdx.x;
  if (idx >= n * co) return;
  int i = idx / co, c = idx - i * co;
  float di   = dis[i];
  float self = 2.f * di * di;
  float a1 = xw1[(size_t)i * co + c] * self;
  float a2 = gated ? xw2[(size_t)i * co + c] * self : 0.f;
  int e0 = rowp[i], e1 = rowp[i + 1];
  for (int e = e0; e < e1; ++e) {
    int ed = csre[e];
    int s  = src[ed];
    float nrm = dis[s] * di;
    a1 += xw1[(size_t)s * co + c] * nrm;
    if (gated) a2 += xw2[(size_t)s * co + c] * nrm;
  }
  float ic = invc[i];
  a1 *= ic;
  float h;
  if (gated) {
    a2 *= ic;
    float el = (a1 > 0.f) ? a1 : expm1f(a1);       // elu
    float sg = 1.f / (1.f + expf(-a2));            // sigmoid
    h = el * sg;
  } else {
    h = 1.f / (1.f + expf(-a1));                   // final sigmoid block
  }
  if (resid) h += resid[idx];
  out[idx] = h;
}

// ---------------- refine: coarse pred -> pixels (seg1) -> scatter-mean into seg2 ----------------
__global__ void refine_scatter_k(const int* __restrict__ seg1, const int* __restrict__ seg2,
                                 const float* __restrict__ out1, float* __restrict__ ssum,
                                 int* __restrict__ rcnt) {
  int idx = blockIdx.x * blockDim.x + threadIdx.x;
  if (idx >= (int)(Bz * HWc)) return;
  int b = idx / (int)HWc;
  int s1 = seg1[idx], s2 = seg2[idx];
  const float* pb = out1 + (size_t)(b * S1c + s1) * 3;
  float* sp = ssum + (size_t)(b * S2c + s2) * 3;
  atomicAdd(sp + 0, pb[0]);
  atomicAdd(sp + 1, pb[1]);
  atomicAdd(sp + 2, pb[2]);
  atomicAdd(&rcnt[b * S2c + s2], 1);
}

__global__ void refine_final_k(const float* __restrict__ ssum, const int* __restrict__ rcnt,
                               const float* __restrict__ x2, const float* __restrict__ mask2,
                               float* __restrict__ x2ref, int n) {
  int i = blockIdx.x * blockDim.x + threadIdx.x;
  if (i >= n) return;
  float cden = fmaxf((float)rcnt[i], 1.f);
  bool use = (mask2[i] == 1.f);
  for (int c = 0; c < 3; ++c) {
    float info = ssum[(size_t)i * 3 + c] / cden;
    x2ref[(size_t)i * 3 + c] = use ? info : x2[(size_t)i * 3 + c];
  }
}

// ---------------- fused pixel gather producing all 9 outputs ----------------
__global__ void gather_out_k(const int* __restrict__ seg1, const int* __restrict__ seg2,
                             const float* __restrict__ x1, const float* __restrict__ gt1,
                             const float* __restrict__ out1, const float* __restrict__ mask1,
                             const float* __restrict__ x2r, const float* __restrict__ gt2,
                             const float* __restrict__ out2, const float* __restrict__ mask2,
                             float* __restrict__ out) {
  int idx = blockIdx.x * blockDim.x + threadIdx.x;
  if (idx >= (int)(Bz * HWc)) return;
  int b = idx / (int)HWc;
  size_t p = (size_t)idx - (size_t)b * HWc;
  int s1 = seg1[idx], s2 = seg2[idx];
  size_t n1i = (size_t)(b * S1c + s1);
  size_t n2i = (size_t)(b * S2c + s2);
  const size_t I3 = (size_t)Bz * 3 * HWc;   // one [B,3,H,W] image
  const size_t I1 = (size_t)Bz * HWc;       // one [B,1,H,W] image
  size_t o = 0;
  for (int c = 0; c < 3; ++c) out[o + ((size_t)b * 3 + c) * HWc + p] = x1[n1i * 3 + c];
  o += I3;  // cin1
  for (int c = 0; c < 3; ++c) out[o + ((size_t)b * 3 + c) * HWc + p] = gt1[n1i * 3 + c];
  o += I3;  // cgt1
  for (int c = 0; c < 3; ++c) out[o + ((size_t)b * 3 + c) * HWc + p] = out1[n1i * 3 + c];
  o += I3;  // cpred1
  out[o + (size_t)b * HWc + p] = mask1[n1i];
  o += I1;  // cmask1
  for (int c = 0; c < 3; ++c) out[o + ((size_t)b * 3 + c) * HWc + p] = x2r[n2i * 3 + c];
  o += I3;  // cin2
  for (int c = 0; c < 3; ++c) out[o + ((size_t)b * 3 + c) * HWc + p] = gt2[n2i * 3 + c];
  o += I3;  // cgt2
  float pr[3];
  for (int c = 0; c < 3; ++c) {
    pr[c] = out2[n2i * 3 + c];
    out[o + ((size_t)b * 3 + c) * HWc + p] = pr[c];
  }
  o += I3;  // cpred2
  out[o + (size_t)b * HWc + p] = mask2[n2i];
  o += I1;  // cmask2
  for (int c = 0; c < 3; ++c) out[o + ((size_t)b * 3 + c) * HWc + p] = pr[c];  // coarse_avg
}

// ---------------- host orchestration ----------------
static inline int cdiv(long a, int b) { return (int)((a + b - 1) / b); }

extern "C" void kernel_launch(void* const* d_in, const int* in_sizes, int n_in,
                              void* d_out, int out_size, void* d_ws, size_t ws_size,
                              hipStream_t stream) {
  const float* x1    = (const float*)d_in[0];
  const float* gt1   = (const float*)d_in[1];
  const float* mask1 = (const float*)d_in[2];
  const float* x2    = (const float*)d_in[3];
  const float* gt2   = (const float*)d_in[4];
  const float* mask2 = (const float*)d_in[5];
  const int* src1 = (const int*)d_in[6];
  const int* dst1 = (const int*)d_in[7];
  const int* src2 = (const int*)d_in[8];
  const int* dst2 = (const int*)d_in[9];
  const int* seg1 = (const int*)d_in[10];
  const int* seg2 = (const int*)d_in[11];
  // last 15 tensors: (w1_0,w2_0, ..., w1_6,w2_6), wf
  const float* Wg[14];
  for (int i = 0; i < 14; ++i) Wg[i] = (const float*)d_in[n_in - 15 + i];
  const float* wf = (const float*)d_in[n_in - 1];

  // ---- workspace carve (floats then ints; all 4B-aligned) ----
  char* ws = (char*)d_ws;
  size_t off = 0;
  auto alloc_f = [&](size_t ne) { float* p = (float*)(ws + off); off += ne * 4; return p; };
  auto alloc_i = [&](size_t ne) { int* p = (int*)(ws + off); off += ne * 4; return p; };
  float* XIN  = alloc_f((size_t)N2c * 4);
  float* H0   = alloc_f((size_t)N2c * 32);
  float* H1   = alloc_f((size_t)N2c * 64);
  float* H2   = alloc_f((size_t)N2c * 128);
  float* FA   = alloc_f((size_t)N2c * 256);
  float* XW1  = alloc_f((size_t)N2c * 256);
  float* XW2  = alloc_f((size_t)N2c * 256);
  float* OUTN1 = alloc_f((size_t)N1c * 3);
  float* OUTN2 = alloc_f((size_t)N2c * 3);
  float* X2REF = alloc_f((size_t)N2c * 3);
  float* DIS  = alloc_f(N2c);
  float* INVC = alloc_f(N2c);
  float* SSUM = alloc_f((size_t)N2c * 3);
  int* CNT  = alloc_i(N2c);
  int* ROWP = alloc_i(N2c + 1);
  int* CURS = alloc_i(N2c);
  int* CSRE = alloc_i(E2c);
  int* RCNT = alloc_i(N2c);

  auto run_net = [&](const float* x, const float* mask, const int* src, const int* dst,
                     int n, int E, float* outN) {
    const int TB = 256;
    build_xin_k<<<cdiv(n, TB), TB, 0, stream>>>(x, mask, XIN, n);
    // degree + CSR (rebuilt per scale; shared by all 8 conv layers)
    zero_i32_k<<<cdiv(n, TB), TB, 0, stream>>>(CNT, n);
    count_deg_k<<<cdiv(E, TB), TB, 0, stream>>>(dst, E, CNT);
    deg_fin_k<<<cdiv(n, TB), TB, 0, stream>>>(CNT, DIS, INVC, n);
    scan_excl_k<<<1, 1024, 0, stream>>>(CNT, ROWP, n);
    copy_i32_k<<<cdiv(n, TB), TB, 0, stream>>>(ROWP, CURS, n);
    csr_place_k<<<cdiv(E, TB), TB, 0, stream>>>(dst, E, CURS, CSRE);
    csr_sort_k<<<cdiv(n, TB), TB, 0, stream>>>(ROWP, CSRE, n);

    const int ci_arr[7] = {4, 32, 64, 128, 256, 128, 64};
    const int co_arr[7] = {32, 64, 128, 256, 128, 64, 32};
    float* outs[7]        = {H0, H1, H2, FA, FA, FA, FA};
    const float* resid[7] = {nullptr, nullptr, nullptr, nullptr, H2, H1, H0};
    const float* in = XIN;
    for (int l = 0; l < 7; ++l) {
      int ci = ci_arr[l], co = co_arr[l];
      int tiles = (n / 16) * (co / 16);
      matmul_wmma_k<<<tiles, 32, 0, stream>>>(in, Wg[2 * l], XW1, n, ci, co);
      matmul_wmma_k<<<tiles, 32, 0, stream>>>(in, Wg[2 * l + 1], XW2, n, ci, co);
      long tot = (long)n * co;
      aggregate_k<<<cdiv(tot, TB), TB, 0, stream>>>(XW1, XW2, ROWP, CSRE, src, DIS, INVC,
                                                    resid[l], outs[l], n, co, 1);
      in = outs[l];
    }
    // final non-gated block: 32 -> 3, sigmoid
    long tot = (long)n * 3;
    matmul_small_k<<<cdiv(tot, TB), TB, 0, stream>>>(in, wf, XW1, n, 32, 3);
    aggregate_k<<<cdiv(tot, TB), TB, 0, stream>>>(XW1, nullptr, ROWP, CSRE, src, DIS, INVC,
                                                  nullptr, outN, n, 3, 0);
  };

  run_net(x1, mask1, src1, dst1, N1c, E1c, OUTN1);
  run_net(x2, mask2, src2, dst2, N2c, E2c, OUTN2);

  // refine: pout -> pixels via seg1, scatter-mean into seg2, masked merge into x2
  const int TB = 256;
  zero_f32_k<<<cdiv((long)N2c * 3, TB), TB, 0, stream>>>(SSUM, N2c * 3);
  zero_i32_k<<<cdiv(N2c, TB), TB, 0, stream>>>(RCNT, N2c);
  refine_scatter_k<<<cdiv((long)(Bz * HWc), TB), TB, 0, stream>>>(seg1, seg2, OUTN1, SSUM, RCNT);
  refine_final_k<<<cdiv(N2c, TB), TB, 0, stream>>>(SSUM, RCNT, x2, mask2, X2REF, N2c);

  // fused gather of all 9 outputs
  gather_out_k<<<cdiv((long)(Bz * HWc), TB), TB, 0, stream>>>(
      seg1, seg2, x1, gt1, OUTN1, mask1, X2REF, gt2, OUTN2, mask2, (float*)d_out);
}